// NgramEnhancer_47837345742857
// MI455X (gfx1250) — compile-verified
//
#include <hip/hip_runtime.h>
#include <math.h>

typedef __attribute__((ext_vector_type(2))) float v2f;
typedef __attribute__((ext_vector_type(8))) float v8f;

#define NWORDS 1024
#define DTOT 2016

__constant__ int c_dims[7] = {32, 64, 128, 256, 512, 512, 512};
__constant__ int c_offs[7] = {0, 32, 96, 224, 480, 992, 1504};

struct LowParams {
  const float* x[7];
  const float* Wl[7];
  const float* bl[7];
  const float* vlw[7];
  const float* vlb[7];
  const float* pos[7];
  const int* cl;
  float* out;
};

struct HighParams {
  const float* Wh[7];
  const float* bh[7];
  const float* Uw;
  const float* Ub;
  const float* vhw;
  const float* vhb;
  float* out;
};

// ---------------------------------------------------------------------------
// Low-level additive attention per (word, ngram-order), fully specialized per
// order so M-tile count / K-chunk count are compile-time constants (keeps the
// WMMA accumulators in fixed registers — no copy storms).
// ---------------------------------------------------------------------------
template <int C, int G, int ORD>
__device__ __forceinline__ void low_body(const LowParams& p, float* xs,
                                         float* sc, float* a_sh) {
  constexpr int MT = (G + 15) >> 4;   // M tiles of 16 gram positions
  constexpr int NCH = C >> 5;         // K chunks of 32 channels
  constexpr int OFFS[7] = {0, 32, 96, 224, 480, 992, 1504};

  const int w = blockIdx.x;
  const int tid = threadIdx.x;
  const int lane = tid & 31;
  const int wv = tid >> 5;      // wave id 0..7
  const int half = lane >> 4;   // 0: lanes 0-15, 1: lanes 16-31
  const int lr = lane & 15;

  const float* xw = p.x[ORD] + (size_t)w * (size_t)(C * G);
  const float* posi = p.pos[ORD] + C;  // rows 1..G of pos table

  if (tid < 64) sc[tid] = 0.0f;

  const bool gw = (wv < 7);        // waves 0..6 own N-tiles (112 >= 100)
  const int n = (wv << 4) + lr;    // output column of this lane
  float vlw_n = 0.0f, bl_n = 0.0f;
  if (gw && n < 100) { vlw_n = p.vlw[ORD][n]; bl_n = p.bl[ORD][n]; }

  v8f acc[MT];
#pragma unroll
  for (int t = 0; t < MT; ++t) acc[t] = (v8f){0, 0, 0, 0, 0, 0, 0, 0};

  for (int kc = 0; kc < NCH; ++kc) {
    const int k0 = kc << 5;
    __syncthreads();  // protect xs reuse
    // Stage x + pos: consecutive tid -> consecutive m (coalesced on x)
#pragma unroll
    for (int it = 0; it < 8; ++it) {
      int idx = tid + it * 256;
      int m = idx & 63;
      int ch = idx >> 6;  // 0..31
      float v = 0.0f;
      if (m < G) v = xw[(size_t)(k0 + ch) * G + m] + posi[(size_t)m * C + (k0 + ch)];
      xs[m * 33 + ch] = v;
    }
    __syncthreads();
    if (gw) {
      // B fragments (Wl chunk) reused across all M-tiles; L2-resident
      v2f bfr[8];
      const float* Wl = p.Wl[ORD];
#pragma unroll
      for (int s = 0; s < 8; ++s) {
        int k = k0 + (s << 2) + half * 2;
        float b0 = 0.0f, b1 = 0.0f;
        if (n < 100) {
          b0 = Wl[(size_t)k * 100 + n];
          b1 = Wl[(size_t)(k + 1) * 100 + n];
        }
        bfr[s] = (v2f){b0, b1};
      }
#pragma unroll
      for (int t = 0; t < MT; ++t) {
#pragma unroll
        for (int s = 0; s < 8; ++s) {
          int kk = (s << 2) + half * 2;
          int m = (t << 4) + lr;
          v2f afr = (v2f){xs[m * 33 + kk], xs[m * 33 + kk + 1]};
          acc[t] = __builtin_amdgcn_wmma_f32_16x16x4_f32(
              false, afr, false, bfr[s], (short)0, acc[t], false, false);
        }
      }
    }
  }

  // sc[m] = sum_n tanh(acc + bl[n]) * vlw[n] : 16-lane reduce + LDS atomics
  if (gw) {
#pragma unroll
    for (int t = 0; t < MT; ++t) {
#pragma unroll
      for (int r = 0; r < 8; ++r) {
        float h = tanhf(acc[t][r] + bl_n);
        float ps = h * vlw_n;
#pragma unroll
        for (int off = 1; off < 16; off <<= 1)
          ps += __shfl_xor(ps, off, 32);
        int m = (t << 4) + r + half * 8;
        if (lr == 0 && m < G) atomicAdd(&sc[m], ps);
      }
    }
  }
  __syncthreads();

  // Masked softmax (matches reference: unmasked max, e*mask, sum + EPS)
  if (tid == 0) {
    int lens = p.cl[w] - ORD;
    if (lens < 0) lens = 0;
    float vlb = p.vlb[ORD][0];
    float mx = -3.402823e38f;
    for (int m = 0; m < G; ++m) {
      float v = sc[m] + vlb;
      sc[m] = v;
      if (v > mx) mx = v;
    }
    float s = 0.0f;
    for (int m = 0; m < G; ++m) {
      float e = (m < lens) ? expf(sc[m] - mx) : 0.0f;
      a_sh[m] = e;
      s += e;
    }
    float inv = 1.0f / (s + 1e-9f);
    for (int m = 0; m < G; ++m) a_sh[m] *= inv;
  }
  __syncthreads();

  // pooled[ch] = sum_m a[m] * x[w, ch, m] (raw x) -> final concat layout
  float* dst = p.out + (size_t)w * DTOT + OFFS[ORD];
  for (int ch = tid; ch < C; ch += 256) {
    const float* xr = xw + (size_t)ch * G;
    float s = 0.0f;
    for (int m = 0; m < G; ++m) s += a_sh[m] * xr[m];
    dst[ch] = s;
  }
}

__global__ __launch_bounds__(256) void ngram_low_attn(LowParams p) {
  __shared__ float xs[64 * 33];
  __shared__ float sc[64];
  __shared__ float a_sh[64];
  switch (blockIdx.y) {
    case 0: low_body<32, 50, 0>(p, xs, sc, a_sh); break;
    case 1: low_body<64, 49, 1>(p, xs, sc, a_sh); break;
    case 2: low_body<128, 48, 2>(p, xs, sc, a_sh); break;
    case 3: low_body<256, 47, 3>(p, xs, sc, a_sh); break;
    case 4: low_body<512, 46, 4>(p, xs, sc, a_sh); break;
    case 5: low_body<512, 45, 5>(p, xs, sc, a_sh); break;
    default: low_body<512, 44, 6>(p, xs, sc, a_sh); break;
  }
}

// ---------------------------------------------------------------------------
// Kernel 2: high-level attention over the 7 pooled vectors, 16 words/block.
// In place on d_out: all pooled reads complete before the scaling writes.
// ---------------------------------------------------------------------------
__global__ __launch_bounds__(256) void ngram_high_attn(HighParams p) {
  const int w0 = blockIdx.x << 4;
  const int tid = threadIdx.x;
  const int lane = tid & 31;
  const int wv = tid >> 5;
  const int half = lane >> 4;
  const int lr = lane & 15;
  const int n = (wv << 4) + lr;  // 0..127 (proj column)

  __shared__ float proj[16 * 137];
  __shared__ float u_sh[16 * 8];
  __shared__ float ah_sh[16 * 8];

  if (tid < 128) u_sh[tid] = 0.0f;

  for (int i = 0; i < 7; ++i) {
    const int c = c_dims[i];
    const float* A = p.out + c_offs[i];  // pooled_i rows, stride DTOT
    const float* B = p.Wh[i];

    // GEMM1: (16 words x c) @ (c x 128)
    v8f acc = (v8f){0, 0, 0, 0, 0, 0, 0, 0};
    for (int k = 0; k < c; k += 4) {
      int ka = k + half * 2;
      const float* arow = A + (size_t)(w0 + lr) * DTOT + ka;
      v2f af = (v2f){arow[0], arow[1]};
      v2f bf = (v2f){B[(size_t)ka * 128 + n], B[(size_t)(ka + 1) * 128 + n]};
      acc = __builtin_amdgcn_wmma_f32_16x16x4_f32(
          false, af, false, bf, (short)0, acc, false, false);
    }
    float bh_n = p.bh[i][n];
    __syncthreads();  // prior iteration's proj reads complete
#pragma unroll
    for (int r = 0; r < 8; ++r) {
      int m = r + half * 8;
      proj[m * 137 + n] = tanhf(acc[r] + bh_n);
    }
    __syncthreads();

    // GEMM2: (16 x 128) @ (128 x 256); each wave owns N-tiles wv and wv+8
    v8f acc2a = (v8f){0, 0, 0, 0, 0, 0, 0, 0};
    v8f acc2b = (v8f){0, 0, 0, 0, 0, 0, 0, 0};
    const int n2a = n, n2b = 128 + n;
#pragma unroll 4
    for (int k = 0; k < 128; k += 4) {
      int ka = k + half * 2;
      v2f af = (v2f){proj[lr * 137 + ka], proj[lr * 137 + ka + 1]};
      v2f ba = (v2f){p.Uw[(size_t)ka * 256 + n2a], p.Uw[(size_t)(ka + 1) * 256 + n2a]};
      v2f bb = (v2f){p.Uw[(size_t)ka * 256 + n2b], p.Uw[(size_t)(ka + 1) * 256 + n2b]};
      acc2a = __builtin_amdgcn_wmma_f32_16x16x4_f32(
          false, af, false, ba, (short)0, acc2a, false, false);
      acc2b = __builtin_amdgcn_wmma_f32_16x16x4_f32(
          false, af, false, bb, (short)0, acc2b, false, false);
    }
    float uba = p.Ub[n2a], ubb = p.Ub[n2b];
    float va = p.vhw[n2a], vb = p.vhw[n2b];
#pragma unroll
    for (int r = 0; r < 8; ++r) {
      float pu = tanhf(acc2a[r] + uba) * va + tanhf(acc2b[r] + ubb) * vb;
#pragma unroll
      for (int off = 1; off < 16; off <<= 1)
        pu += __shfl_xor(pu, off, 32);
      int m = r + half * 8;
      if (lr == 0) atomicAdd(&u_sh[m * 8 + i], pu);
    }
    __syncthreads();
  }
  __syncthreads();

  // Softmax over the 7 orders, one word per thread
  if (tid < 16) {
    float vhb = p.vhb[0];
    float mx = -3.402823e38f;
#pragma unroll
    for (int i = 0; i < 7; ++i) {
      float v = u_sh[tid * 8 + i] + vhb;
      u_sh[tid * 8 + i] = v;
      if (v > mx) mx = v;
    }
    float s = 0.0f;
#pragma unroll
    for (int i = 0; i < 7; ++i) {
      float e = expf(u_sh[tid * 8 + i] - mx);
      ah_sh[tid * 8 + i] = e;
      s += e;
    }
    float inv = 1.0f / (s + 1e-9f);
#pragma unroll
    for (int i = 0; i < 7; ++i) ah_sh[tid * 8 + i] *= inv;
  }
  __syncthreads();

  // Scale the 16 x 2016 output rows in place
  for (int idx = tid; idx < 16 * DTOT; idx += 256) {
    int m = idx / DTOT;
    int chn = idx - m * DTOT;
    int oi = 0;
#pragma unroll
    for (int j = 1; j < 7; ++j)
      if (chn >= c_offs[j]) oi = j;
    size_t o = (size_t)(w0 + m) * DTOT + chn;
    p.out[o] *= ah_sh[m * 8 + oi];
  }
}

extern "C" void kernel_launch(void* const* d_in, const int* in_sizes, int n_in,
                              void* d_out, int out_size, void* d_ws, size_t ws_size,
                              hipStream_t stream) {
  (void)in_sizes; (void)n_in; (void)out_size; (void)d_ws; (void)ws_size;
  // Input order (setup_inputs dict order, lists flattened):
  // 0-6: x0..x6 | 7: char_lengths | 8-14: Wl | 15-21: bl | 22-28: vlw
  // 29-35: vlb | 36-42: pos | 43-49: Wh | 50-56: bh | 57: Uw | 58: Ub
  // 59: vhw | 60: vhb
  LowParams lp;
  for (int i = 0; i < 7; ++i) {
    lp.x[i]   = (const float*)d_in[i];
    lp.Wl[i]  = (const float*)d_in[8 + i];
    lp.bl[i]  = (const float*)d_in[15 + i];
    lp.vlw[i] = (const float*)d_in[22 + i];
    lp.vlb[i] = (const float*)d_in[29 + i];
    lp.pos[i] = (const float*)d_in[36 + i];
  }
  lp.cl = (const int*)d_in[7];
  lp.out = (float*)d_out;

  HighParams hp;
  for (int i = 0; i < 7; ++i) {
    hp.Wh[i] = (const float*)d_in[43 + i];
    hp.bh[i] = (const float*)d_in[50 + i];
  }
  hp.Uw  = (const float*)d_in[57];
  hp.Ub  = (const float*)d_in[58];
  hp.vhw = (const float*)d_in[59];
  hp.vhb = (const float*)d_in[60];
  hp.out = (float*)d_out;

  dim3 grid1(NWORDS, 7);
  ngram_low_attn<<<grid1, 256, 0, stream>>>(lp);
  ngram_high_attn<<<NWORDS / 16, 256, 0, stream>>>(hp);
}